// L1Distance_31739808318036
// MI455X (gfx1250) — compile-verified
//
#include <hip/hip_runtime.h>
#include <cstdint>

// Pairwise L1 distance, N=M=2048, D=64.
// res[i,j] = sum_d |x1[i,d] - x2[j,d]|   (mean-centering cancels exactly)
//
// 64x64 output tile per 256-thread block; 4x4 register tile per thread.
// Tiles are stored TRANSPOSED in LDS ([d][point]) so the inner loop is two
// conflict-free ds_load_b128 per d. The transpose is done for free by the
// CDNA5 async global->LDS engine (per-lane LDS destination addresses).

__global__ __launch_bounds__(256) void l1dist_kernel(
    const float* __restrict__ x1,
    const float* __restrict__ x2,
    float* __restrict__ out,
    int M /* row stride of out */) {
  __shared__ float As[64][64];  // [d][n-in-tile]
  __shared__ float Bs[64][64];  // [d][m-in-tile]

  const int tid = threadIdx.x;
  const int tx = tid & 15;   // column group: cols tx*4 .. tx*4+3
  const int ty = tid >> 4;   // row group:    rows ty*4 .. ty*4+3
  const int n0 = blockIdx.y * 64;
  const int m0 = blockIdx.x * 64;

  // Tile base pointers (row-major [point][d], 64 floats per point).
  const float* aSrc = x1 + (size_t)n0 * 64;
  const float* bSrc = x2 + (size_t)m0 * 64;

  // LDS byte offsets of the shared arrays (generic LDS addr: low 32 bits are
  // the in-allocation offset).
  const uint32_t aBase = (uint32_t)(uintptr_t)(&As[0][0]);
  const uint32_t bBase = (uint32_t)(uintptr_t)(&Bs[0][0]);

  // Async fill with on-the-fly transpose: flat element l = k*256 + tid reads
  // global offset l*4 (coalesced) and lands at LDS [d][n] = [(l&63)][(l>>6)].
#pragma unroll
  for (int k = 0; k < 16; ++k) {
    const uint32_t l = (uint32_t)(k * 256 + tid);
    const uint32_t gOff = l << 2;                         // global byte offset
    const uint32_t pnt = l >> 6;                          // point index in tile
    const uint32_t dIdx = l & 63u;                        // d index
    const uint32_t tOff = ((dIdx << 6) + pnt) << 2;       // transposed LDS byte off
    const uint32_t lA = aBase + tOff;
    const uint32_t lB = bBase + tOff;
    asm volatile("global_load_async_to_lds_b32 %0, %1, %2"
                 :: "v"(lA), "v"(gOff), "s"(aSrc) : "memory");
    asm volatile("global_load_async_to_lds_b32 %0, %1, %2"
                 :: "v"(lB), "v"(gOff), "s"(bSrc) : "memory");
  }
  asm volatile("s_wait_asynccnt 0" ::: "memory");
  __syncthreads();

  float acc[4][4] = {{0.f, 0.f, 0.f, 0.f},
                     {0.f, 0.f, 0.f, 0.f},
                     {0.f, 0.f, 0.f, 0.f},
                     {0.f, 0.f, 0.f, 0.f}};

#pragma unroll 8
  for (int d = 0; d < 64; ++d) {
    // a: 4 rows at this d (broadcast across tx lanes); b: 4 cols (64 distinct
    // banks across the wave) -- both single ds_load_b128, conflict-free.
    const float4 a = *reinterpret_cast<const float4*>(&As[d][ty * 4]);
    const float4 b = *reinterpret_cast<const float4*>(&Bs[d][tx * 4]);
    const float ar[4] = {a.x, a.y, a.z, a.w};
    const float bc[4] = {b.x, b.y, b.z, b.w};
#pragma unroll
    for (int r = 0; r < 4; ++r) {
#pragma unroll
      for (int c = 0; c < 4; ++c) {
        // v_sub_f32 + v_add_f32 with |src| modifier: 2 VALU per element.
        acc[r][c] += __builtin_fabsf(ar[r] - bc[c]);
      }
    }
  }

  // Coalesced b128 stores: lanes tx=0..15 cover 64 consecutive floats per row.
#pragma unroll
  for (int r = 0; r < 4; ++r) {
    const int row = n0 + ty * 4 + r;
    float4 v;
    v.x = fmaxf(acc[r][0], 0.f);
    v.y = fmaxf(acc[r][1], 0.f);
    v.z = fmaxf(acc[r][2], 0.f);
    v.w = fmaxf(acc[r][3], 0.f);
    *reinterpret_cast<float4*>(&out[(size_t)row * M + m0 + tx * 4]) = v;
  }
}

extern "C" void kernel_launch(void* const* d_in, const int* in_sizes, int n_in,
                              void* d_out, int out_size, void* d_ws, size_t ws_size,
                              hipStream_t stream) {
  (void)n_in; (void)d_ws; (void)ws_size; (void)out_size;
  const float* x1 = (const float*)d_in[0];
  const float* x2 = (const float*)d_in[1];
  float* out = (float*)d_out;

  const int D = 64;
  const int N = in_sizes[0] / D;  // 2048
  const int M = in_sizes[1] / D;  // 2048

  dim3 grid(M / 64, N / 64);
  dim3 block(256);
  l1dist_kernel<<<grid, block, 0, stream>>>(x1, x2, out, M);
}